// SupernodeEncoder_71116068487360
// MI455X (gfx1250) — compile-verified
//
#include <hip/hip_runtime.h>

// ---------------------------------------------------------------------------
// SupernodeEncoder for MI455X (gfx1250, wave32)
//
// Roofline: projection writes 51.2 MB (bandwidth bound, ~2.2us @ 23.3TB/s);
// x then stays L2-resident (51.2MB < 192MB L2) for the gather phase.
// Ball-query mask is ~0.05% dense (~26 neighbors / 50000 points), so the
// aggregation is sparse; the only dense math is the MxN distance cross-term,
// computed 16x16 tiles at a time with V_WMMA_F32_16X16X4_F32 (K=3 pad to 4).
// The N dimension is split into 8 segments for occupancy (512 blocks), with
// per-block LDS partials flushed via global f32 atomics and a finalize pass.
// ---------------------------------------------------------------------------

typedef __attribute__((ext_vector_type(2))) float v2f;
typedef __attribute__((ext_vector_type(8))) float v8f;

#define NPTS    50000
#define MSUP    1024
#define CDIM    256
#define RADIUS2 0.0025f        // 0.05^2
#define NCHUNKS (NPTS / 16)    // 3125 exactly
#define SEG     8              // point-range segments (grid.y)

// ---------------------------------------------------------------------------
// Kernel 0: zero the output accumulator and global counts (harness poisons).
// ---------------------------------------------------------------------------
__global__ __launch_bounds__(256) void init_kernel(float* __restrict__ out,
                                                   float* __restrict__ cnt_ws)
{
  int idx = blockIdx.x * 256 + threadIdx.x;
  if (idx < MSUP * CDIM) out[idx] = 0.0f;
  if (idx < MSUP)        cnt_ws[idx] = 0.0f;
}

// ---------------------------------------------------------------------------
// Kernel 1: x[p, c] = b[c] + sum_k concat(pos,fun)[p,k] * W[c,k]
// 256 threads (one channel each) x 16 points; W row in registers, inputs in
// LDS, fully coalesced 1KB-per-point output writes.
// ---------------------------------------------------------------------------
__global__ __launch_bounds__(256) void proj_kernel(
    const float* __restrict__ pos, const float* __restrict__ fun,
    const float* __restrict__ W, const float* __restrict__ b,
    float* __restrict__ x)
{
  __shared__ float inp[16][8];
  const int c  = threadIdx.x;
  const int p0 = blockIdx.x * 16;

  if (threadIdx.x < 16 * 8) {
    int p = threadIdx.x >> 3;
    int k = threadIdx.x & 7;
    inp[p][k] = (k < 3) ? pos[(p0 + p) * 3 + k] : fun[(p0 + p) * 5 + (k - 3)];
  }

  float w[8];
  #pragma unroll
  for (int k = 0; k < 8; ++k) w[k] = W[c * 8 + k];
  const float bias = b[c];

  __syncthreads();

  #pragma unroll 4
  for (int p = 0; p < 16; ++p) {
    float acc = bias;
    #pragma unroll
    for (int k = 0; k < 8; ++k) acc = fmaf(inp[p][k], w[k], acc);
    x[(size_t)(p0 + p) * CDIM + c] = acc;
  }
}

// ---------------------------------------------------------------------------
// Kernel 2: block = (16-supernode tile, point segment). 8 waves stride over
// 16-point chunks. Per chunk: WMMA f32 16x16x4 gives the 16x16 sp.p tile,
// d2 = |sp|^2 + |p|^2 - 2 dot, ballot the hits (rare), wave-gather x[p,:]
// and ds_add_f32 into the LDS tile. Point loads are software-pipelined one
// chunk ahead to hide L2 latency. Flush via global f32 atomics.
// ---------------------------------------------------------------------------
__global__ __launch_bounds__(256) void agg_kernel(
    const float* __restrict__ pos, const int* __restrict__ sidx,
    const float* __restrict__ x, float* __restrict__ out,
    float* __restrict__ cnt_ws)
{
  __shared__ float sp[16][4];          // supernode x,y,z,|sp|^2
  __shared__ float latent[16][CDIM];   // 16 KB accumulator tile
  __shared__ float cnt[16];

  const int tid  = threadIdx.x;
  const int lane = tid & 31;
  const int wave = tid >> 5;

  if (tid < 16) {
    int si = sidx[blockIdx.x * 16 + tid];
    float px = pos[si * 3 + 0], py = pos[si * 3 + 1], pz = pos[si * 3 + 2];
    sp[tid][0] = px; sp[tid][1] = py; sp[tid][2] = pz;
    sp[tid][3] = px * px + py * py + pz * pz;
    cnt[tid] = 0.0f;
  }
  #pragma unroll
  for (int m = 0; m < 16; ++m) latent[m][tid] = 0.0f;
  __syncthreads();

  // --- A matrix (16x4 f32, ISA layout: VGPR0 holds K=0|K=2, VGPR1 K=1|K=3,
  //     M = lane % 16, K half selected by lane >= 16) -----------------------
  const bool hi = lane >= 16;
  const int  nl = lane & 15;
  v2f a;
  a.x = hi ? sp[nl][2] : sp[nl][0];   // K=2 (z) : K=0 (x)
  a.y = hi ? 0.0f      : sp[nl][1];   // K=3 (pad) : K=1 (y)

  // |sp|^2 aligned to the D-tile layout: VGPR r covers rows r and r+8.
  float snorm[8];
  #pragma unroll
  for (int r = 0; r < 8; ++r) snorm[r] = sp[r + (hi ? 8 : 0)][3];

  // Chunk range for this segment; waves stride by 8 within it.
  const int c0 = (int)(((long)blockIdx.y     * NCHUNKS) / SEG);
  const int c1 = (int)(((long)(blockIdx.y+1) * NCHUNKS) / SEG);

  int chunk = c0 + wave;
  float px = 0.0f, py = 0.0f, pz = 0.0f;
  if (chunk < c1) {                                  // prologue prefetch
    const float* pp = pos + (size_t)(chunk * 16 + nl) * 3;
    px = pp[0]; py = pp[1]; pz = pp[2];
  }

  while (chunk < c1) {
    const int j0 = chunk * 16;
    const int nchunk = chunk + 8;
    float nx = 0.0f, ny = 0.0f, nz = 0.0f;
    if (nchunk < c1) {                               // prefetch next chunk
      const float* pp = pos + (size_t)(nchunk * 16 + nl) * 3;
      nx = pp[0]; ny = pp[1]; nz = pp[2];
    }

    const float pn2 = px * px + py * py + pz * pz;

    // B matrix (4x16 f32): row K striped across a lane half per VGPR.
    v2f bm;
    bm.x = hi ? pz   : px;            // K=2 : K=0
    bm.y = hi ? 0.0f : py;            // K=3 : K=1

    v8f dot = {};
    dot = __builtin_amdgcn_wmma_f32_16x16x4_f32(
        /*neg_a=*/false, a, /*neg_b=*/false, bm,
        /*c_mod=*/(short)0, dot, /*reuse_a=*/false, /*reuse_b=*/false);

    #pragma unroll
    for (int r = 0; r < 8; ++r) {
      const float d2 = snorm[r] + pn2 - 2.0f * dot[r];
      unsigned mask = __builtin_amdgcn_ballot_w32(d2 <= RADIUS2);
      while (mask) {                  // wave-uniform; expected ~0.13 hits/chunk
        const int bl = __ffs(mask) - 1;
        mask &= mask - 1;
        const int M    = r + ((bl >= 16) ? 8 : 0);
        const int pidx = j0 + (bl & 15);
        const float* xr = x + (size_t)pidx * CDIM;
        #pragma unroll
        for (int q = 0; q < 8; ++q) { // 32 lanes x 8 channels = 256
          const int c = lane * 8 + q;
          atomicAdd(&latent[M][c], xr[c]);   // ds_add_f32
        }
        if (lane == 0) atomicAdd(&cnt[M], 1.0f);
      }
    }

    px = nx; py = ny; pz = nz;
    chunk = nchunk;
  }
  __syncthreads();

  // Flush block partials to global accumulators.
  #pragma unroll
  for (int m = 0; m < 16; ++m)
    atomicAdd(&out[((size_t)blockIdx.x * 16 + m) * CDIM + tid], latent[m][tid]);
  if (tid < 16)
    atomicAdd(&cnt_ws[blockIdx.x * 16 + tid], cnt[tid]);
}

// ---------------------------------------------------------------------------
// Kernel 3: mean-normalize, coalesced (thread = channel, loop over rows).
// ---------------------------------------------------------------------------
__global__ __launch_bounds__(256) void finalize_kernel(
    const float* __restrict__ cnt_ws, float* __restrict__ out)
{
  const int tid = threadIdx.x;
  #pragma unroll
  for (int m = 0; m < 16; ++m) {
    float c = cnt_ws[blockIdx.x * 16 + m];
    c = (c < 1.0f) ? 1.0f : c;
    const size_t o = ((size_t)blockIdx.x * 16 + m) * CDIM + tid;
    out[o] = out[o] / c;
  }
}

// ---------------------------------------------------------------------------
extern "C" void kernel_launch(void* const* d_in, const int* in_sizes, int n_in,
                              void* d_out, int out_size, void* d_ws, size_t ws_size,
                              hipStream_t stream) {
  const float* pos  = (const float*)d_in[0];   // [N,3]
  const float* fun  = (const float*)d_in[1];   // [N,5]
  const int*   sidx = (const int*)d_in[2];     // [M]
  const float* W    = (const float*)d_in[3];   // [C,8]
  const float* b    = (const float*)d_in[4];   // [C]
  float* out = (float*)d_out;                  // [M,C]

  float* x      = (float*)d_ws;                       // [N,C] = 51.2 MB
  float* cnt_ws = (float*)d_ws + (size_t)NPTS * CDIM; // [M]   = 4 KB

  init_kernel<<<(MSUP * CDIM + 255) / 256, 256, 0, stream>>>(out, cnt_ws);
  proj_kernel<<<NPTS / 16, 256, 0, stream>>>(pos, fun, W, b, x);
  agg_kernel<<<dim3(MSUP / 16, SEG), 256, 0, stream>>>(pos, sidx, x, out, cnt_ws);
  finalize_kernel<<<MSUP / 16, 256, 0, stream>>>(cnt_ws, out);
}